// ScaledDotProductAttention_13692355740492
// MI455X (gfx1250) — compile-verified
//
#include <hip/hip_runtime.h>
#include <hip/hip_bf16.h>
#include <math.h>

typedef float v2f __attribute__((ext_vector_type(2)));
typedef float v8f __attribute__((ext_vector_type(8)));

#define BATCH 128
#define LEN   512
#define DKDIM 64
#define INV_TEMPER 0.125f
#define POSS 532   // pos_lds row stride (f32): 16 rows; 8*532 % 64 == 32 -> conflict-free C-store
#define PS   516   // p_lds   row stride (f32): 516 % 64 == 4 -> conflict-free A-loads & C-store

__global__ __launch_bounds__(128, 1) void dpa_attn_kernel(
    const float* __restrict__ q, const float* __restrict__ k,
    const float* __restrict__ v, const float* __restrict__ pos,
    const unsigned char* __restrict__ mask,
    float* __restrict__ out, float* __restrict__ attn_out)
{
  __shared__ float pos_lds[16 * POSS];  // positional scores, local cols [0, 528)
  __shared__ float p_lds[16 * PS];      // scores -> probabilities, cols [0, 512)

  const int tid   = threadIdx.x;
  const int wave  = tid >> 5;
  const int lane  = tid & 31;
  const int lhalf = lane >> 4;   // 0 or 1 (selects K-pair / M-half per ISA layout)
  const int l16   = lane & 15;
  const int b     = blockIdx.x >> 5;        // 32 q-tiles per batch
  const int q0    = (blockIdx.x & 31) << 4; // first q row of tile

  // ---- Q tile in A-matrix layout: 16 chained 16x4 chunks cover K=64.
  // Lane m<16 holds row q0+m, VGPR pair = K {4s, 4s+1}; lane m+16 holds K {4s+2, 4s+3}.
  v2f qa[16];
  {
    const float* qrow = q + (size_t)(b * LEN + q0 + l16) * DKDIM;
#pragma unroll
    for (int s = 0; s < 16; ++s)
      qa[s] = *(const v2f*)(qrow + 4 * s + 2 * lhalf);
  }

  // ================= Phase A: positional scores ==========================
  // bias[i][j] = (Q[q0+i] . P[q0+i+j]) / 8 ; need pos cols q0 .. q0+526 -> 33 tiles.
  for (int t = wave; t < 33; t += 4) {
    const int col = q0 + t * 16 + l16;                 // <= 1023 by construction
    const float* prow = pos + (size_t)(b * (2 * LEN) + col) * DKDIM;
    v8f acc = {};
#pragma unroll
    for (int s = 0; s < 16; ++s) {
      v2f bb = *(const v2f*)(prow + 4 * s + 2 * lhalf);  // B[k][n] = P[col n][k]
      acc = __builtin_amdgcn_wmma_f32_16x16x4_f32(false, qa[s], false, bb,
                                                  (short)0, acc, false, false);
    }
#pragma unroll
    for (int r = 0; r < 8; ++r) {
      const int m = r + (lhalf << 3);                  // C layout: M = r (+8 for hi half)
      pos_lds[m * POSS + t * 16 + l16] = acc[r] * INV_TEMPER;
    }
  }
  __syncthreads();

  // ================= Phase B: masked, biased scores ======================
  for (int t = wave; t < 32; t += 4) {
    const int n0 = t * 16;
    const float* krow = k + (size_t)(b * LEN + n0 + l16) * DKDIM;
    v8f acc = {};
#pragma unroll
    for (int s = 0; s < 16; ++s) {
      v2f bb = *(const v2f*)(krow + 4 * s + 2 * lhalf);
      acc = __builtin_amdgcn_wmma_f32_16x16x4_f32(false, qa[s], false, bb,
                                                  (short)0, acc, false, false);
    }
#pragma unroll
    for (int r = 0; r < 8; ++r) {
      const int m = r + (lhalf << 3);
      const int n = n0 + l16;
      float sc = acc[r] * INV_TEMPER + pos_lds[m * POSS + m + n]; // stripe gather
      if (mask[(size_t)(b * LEN + q0 + m) * LEN + n]) sc = -INFINITY;
      p_lds[m * PS + n] = sc;
    }
  }
  __syncthreads();

  // ================= Softmax over key axis (8 threads / row) ============
  {
    const int row = tid >> 3;
    const int sub = tid & 7;
    float* prow = p_lds + row * PS;
    float mx = -INFINITY;
    for (int j = sub; j < LEN; j += 8) mx = fmaxf(mx, prow[j]);
#pragma unroll
    for (int d = 1; d < 8; d <<= 1) mx = fmaxf(mx, __shfl_xor(mx, d, 8));
    float sum = 0.0f;
    for (int j = sub; j < LEN; j += 8) {
      const float e = expf(prow[j] - mx);  // exp(-inf)=0 for masked entries
      prow[j] = e;
      sum += e;
    }
#pragma unroll
    for (int d = 1; d < 8; d <<= 1) sum += __shfl_xor(sum, d, 8);
    const float inv = 1.0f / sum;
    for (int j = sub; j < LEN; j += 8) prow[j] *= inv;
  }
  __syncthreads();

  // ---- write attn tile (coalesced: consecutive tids -> consecutive cols)
  {
    float* abase = attn_out + (size_t)(b * LEN + q0) * LEN;
    for (int idx = tid; idx < 16 * LEN; idx += 128) {
      const int rrow = idx >> 9;
      const int ccol = idx & (LEN - 1);
      abase[idx] = p_lds[rrow * PS + ccol];
    }
  }

  // ================= Phase C: output = P . V  (wave w -> dims 16w..16w+15)
  {
    const int d0 = wave << 4;
    v8f acc = {};
    for (int ks = 0; ks < 128; ++ks) {
      const int k0 = ks << 2;
      // A: P rows from LDS (conflict-free b64)
      v2f pa = *(const v2f*)(p_lds + l16 * PS + k0 + 2 * lhalf);
      // B: B[kk][n] = V[k0+kk][d0+n]; lane half selects key pair
      const float* vcol = v + (size_t)(b * LEN + k0 + 2 * lhalf) * DKDIM + d0 + l16;
      v2f bb;
      bb.x = vcol[0];
      bb.y = vcol[DKDIM];
      acc = __builtin_amdgcn_wmma_f32_16x16x4_f32(false, pa, false, bb,
                                                  (short)0, acc, false, false);
    }
#pragma unroll
    for (int r = 0; r < 8; ++r) {
      const int m = r + (lhalf << 3);
      out[(size_t)(b * LEN + q0 + m) * DKDIM + d0 + l16] = acc[r];
    }
  }
}

extern "C" void kernel_launch(void* const* d_in, const int* in_sizes, int n_in,
                              void* d_out, int out_size, void* d_ws, size_t ws_size,
                              hipStream_t stream) {
  (void)in_sizes; (void)n_in; (void)out_size; (void)d_ws; (void)ws_size;
  const float* q   = (const float*)d_in[0];
  const float* k   = (const float*)d_in[1];
  const float* v   = (const float*)d_in[2];
  const float* pos = (const float*)d_in[3];
  const unsigned char* mask = (const unsigned char*)d_in[4];  // jnp bool -> 1 byte
  float* out  = (float*)d_out;
  float* attn = out + (size_t)BATCH * LEN * DKDIM;  // tuple order: (output, attn)

  dim3 grid(BATCH * (LEN / 16));  // 4096 workgroups
  dim3 block(128);                // 4 wave32s
  hipLaunchKernelGGL(dpa_attn_kernel, grid, block, 0, stream,
                     q, k, v, pos, mask, out, attn);
}